// RotEquiAttention_19000935318152
// MI455X (gfx1250) — compile-verified
//
#include <hip/hip_runtime.h>
#include <math.h>

typedef float v2f __attribute__((ext_vector_type(2)));
typedef float v8f __attribute__((ext_vector_type(8)));

#define B_    8
#define C_    256
#define H_    128
#define W_    128
#define HW_   (H_ * W_)
#define MID_  64
#define NANG_ 4
#define BNE_  1e-5f

// ---------------------------------------------------------------------------
// Per-angle sampling tables (vectorized int4/float4 entries).
// Forward grid (c,s): F_off[pix] = 4 clipped src indices, F_w = 4 bilinear wts.
// Also scatter-adds Wmap (forward bilinear weights / HW) for the pooling pass.
// Inverse grid (c,-s): I_idx[pix] = 4 dest-pixel corner indices, I_w = 4 wts.
// ---------------------------------------------------------------------------
__device__ __forceinline__ void bilin4(float gx, float gy, int* offs, float* ws) {
    float ix = (gx + 1.f) * 0.5f * (float)(W_ - 1);
    float iy = (gy + 1.f) * 0.5f * (float)(H_ - 1);
    float x0 = floorf(ix), y0 = floorf(iy);
    float x1 = x0 + 1.f,  y1 = y0 + 1.f;
    float wx1 = ix - x0, wx0 = 1.f - wx1;
    float wy1 = iy - y0, wy0 = 1.f - wy1;
    float xs[2] = {x0, x1}, ys[2] = {y0, y1};
    float wxs[2] = {wx0, wx1}, wys[2] = {wy0, wy1};
#pragma unroll
    for (int jy = 0; jy < 2; ++jy)
#pragma unroll
        for (int jx = 0; jx < 2; ++jx) {
            float xf = xs[jx], yf = ys[jy];
            bool valid = (xf >= 0.f) && (xf <= (float)(W_ - 1)) &&
                         (yf >= 0.f) && (yf <= (float)(H_ - 1));
            int xi = (int)fminf(fmaxf(xf, 0.f), (float)(W_ - 1));
            int yi = (int)fminf(fmaxf(yf, 0.f), (float)(H_ - 1));
            int k = jy * 2 + jx;
            offs[k] = yi * W_ + xi;
            ws[k]   = wxs[jx] * wys[jy] * (valid ? 1.f : 0.f);
        }
}

__global__ void zero_wmap_kernel(float* __restrict__ Wmap) {
    int t = blockIdx.x * blockDim.x + threadIdx.x;
    if (t < HW_) Wmap[t] = 0.f;
}

__global__ void build_tables_kernel(const float* __restrict__ angles, int ang,
                                    int4* __restrict__ F_off, float4* __restrict__ F_w,
                                    int4* __restrict__ I_idx, float4* __restrict__ I_w,
                                    float* __restrict__ Wmap) {
    int t = blockIdx.x * blockDim.x + threadIdx.x;
    if (t >= HW_) return;
    int y = t / W_, x = t % W_;
    float a  = angles[ang];
    float cs = cosf(a), sn = sinf(a);
    float X = -1.f + (float)x * (2.f / (float)(W_ - 1));
    float Y = -1.f + (float)y * (2.f / (float)(H_ - 1));

    int offs[4]; float ws[4];
    // forward rotation
    bilin4(cs * X - sn * Y, sn * X + cs * Y, offs, ws);
    F_off[t] = make_int4(offs[0], offs[1], offs[2], offs[3]);
    F_w[t]   = make_float4(ws[0], ws[1], ws[2], ws[3]);
#pragma unroll
    for (int k = 0; k < 4; ++k)
        if (ws[k] != 0.f) atomicAdd(&Wmap[offs[k]], ws[k] * (1.f / (float)HW_));
    // inverse rotation (c, -s)
    bilin4(cs * X + sn * Y, -sn * X + cs * Y, offs, ws);
    I_idx[t] = make_int4(offs[0], offs[1], offs[2], offs[3]);
    I_w[t]   = make_float4(ws[0], ws[1], ws[2], ws[3]);
}

// ---------------------------------------------------------------------------
// p[b,c] = mean_{H,W}(rotate(x)) = dot(x plane, Wmap). Streaming, coalesced.
// ---------------------------------------------------------------------------
__global__ __launch_bounds__(256) void pool_kernel(const float* __restrict__ x,
                                                   const float* __restrict__ Wmap,
                                                   float* __restrict__ p) {
    __shared__ float red[256];
    int bc = blockIdx.x;  // b*C_ + c
    const float* plane = x + (size_t)bc * HW_;
    float acc = 0.f;
    for (int i = threadIdx.x; i < HW_; i += 256) {
        __builtin_prefetch(plane + i + 4096, 0, 1);  // global_prefetch_b8
        acc += plane[i] * Wmap[i];
    }
    red[threadIdx.x] = acc;
    __syncthreads();
    for (int s = 128; s > 0; s >>= 1) {
        if (threadIdx.x < (unsigned)s) red[threadIdx.x] += red[threadIdx.x + s];
        __syncthreads();
    }
    if (threadIdx.x == 0) p[bc] = red[0];
}

// ---------------------------------------------------------------------------
// MLP: h = relu(BN(p @ w1^T)); ca = sigmoid(h @ w2^T)  -- via WMMA f32 16x16x4.
// One workgroup, 16 waves. p is staged zero-padded into LDS so the WMMA K-loop
// has NO per-lane predication (unconditional ds_load_b64 + global_load_b64).
// A layout (16x4 f32): lanes 0-15 M=lane {K0,K1}; lanes 16-31 M=lane-16 {K2,K3}.
// B layout (4x16 f32): lanes 0-15 N=lane {K0,K1}; lanes 16-31 N=lane-16 {K2,K3}.
// D layout: VGPR i, lanes 0-15 -> (M=i, N=lane); lanes 16-31 -> (M=8+i, N=lane-16).
// ---------------------------------------------------------------------------
__global__ __launch_bounds__(512) void fc_kernel(const float* __restrict__ p,
                                                 const float* __restrict__ w1,
                                                 const float* __restrict__ bn_gamma,
                                                 const float* __restrict__ bn_beta,
                                                 const float* __restrict__ bn_mean,
                                                 const float* __restrict__ bn_var,
                                                 const float* __restrict__ w2,
                                                 float* __restrict__ ca) {
    __shared__ float p_lds[16][C_];   // rows 8..15 zero-padded (16 KB)
    __shared__ float h_lds[16][MID_]; // rows 8..15 zero-padded (4 KB)
    int tid   = threadIdx.x;
    int wave  = tid >> 5;
    int lane  = tid & 31;
    int row   = lane & 15;   // M (A) or N (B) within tile
    int khalf = lane >> 4;   // 0: K0/K1, 1: K2/K3

    // stage p (2048 valid floats) + zero padding, fully unconditional afterwards
    for (int i = tid; i < 16 * C_; i += 512)
        (&p_lds[0][0])[i] = (i < B_ * C_) ? p[i] : 0.f;
    __syncthreads();

    if (wave < 4) {  // FC1: h = p @ w1^T, 4 tiles of 16 cols, K = 256
        int ntile = wave;
        const float* w1r = w1 + (size_t)(ntile * 16 + row) * C_;
        v8f acc = {0.f, 0.f, 0.f, 0.f, 0.f, 0.f, 0.f, 0.f};
        for (int k0 = 0; k0 < C_; k0 += 4) {
            v2f av = *(const v2f*)&p_lds[row][k0 + 2 * khalf];
            v2f bv = *(const v2f*)&w1r[k0 + 2 * khalf];
            acc = __builtin_amdgcn_wmma_f32_16x16x4_f32(
                false, av, false, bv, (short)0, acc, false, false);
        }
#pragma unroll
        for (int i = 0; i < 8; ++i) {
            int M   = i + 8 * khalf;
            int col = ntile * 16 + row;
            float v   = acc[i];
            float bnv = (v - bn_mean[col]) * rsqrtf(bn_var[col] + BNE_) *
                        bn_gamma[col] + bn_beta[col];
            h_lds[M][col] = (M < B_) ? fmaxf(bnv, 0.f) : 0.f;
        }
    }
    __syncthreads();

    {   // FC2: ca = sigmoid(h @ w2^T), 16 tiles of 16 cols, K = 64
        int ntile = wave;
        const float* w2r = w2 + (size_t)(ntile * 16 + row) * MID_;
        v8f acc = {0.f, 0.f, 0.f, 0.f, 0.f, 0.f, 0.f, 0.f};
        for (int k0 = 0; k0 < MID_; k0 += 4) {
            v2f av = *(const v2f*)&h_lds[row][k0 + 2 * khalf];
            v2f bv = *(const v2f*)&w2r[k0 + 2 * khalf];
            acc = __builtin_amdgcn_wmma_f32_16x16x4_f32(
                false, av, false, bv, (short)0, acc, false, false);
        }
        if (lane < 16) {  // rows M=0..7 live in VGPRs 0..7, lanes 0..15
#pragma unroll
            for (int i = 0; i < B_; ++i) {
                float s = 1.f / (1.f + expf(-acc[i]));
                ca[i * C_ + ntile * 16 + lane] = s;
            }
        }
    }
}

// ---------------------------------------------------------------------------
// cat[b,{avg,max},y,x] over channels of fca = rotate(x)*ca  (table gather).
// ---------------------------------------------------------------------------
__global__ __launch_bounds__(256) void cat_kernel(const float* __restrict__ x,
                                                  const float* __restrict__ ca,
                                                  const int4* __restrict__ F_off,
                                                  const float4* __restrict__ F_w,
                                                  float* __restrict__ cat) {
    int t = blockIdx.x * blockDim.x + threadIdx.x;
    if (t >= B_ * HW_) return;
    int b = t / HW_, pix = t % HW_;
    int4   o = F_off[pix];
    float4 w = F_w[pix];
    const float* xb  = x + (size_t)b * C_ * HW_;
    const float* cab = ca + b * C_;
    float sum = 0.f, mx = -INFINITY;
    for (int c = 0; c < C_; ++c) {
        const float* pl = xb + (size_t)c * HW_;
        float v = w.x * pl[o.x] + w.y * pl[o.y] + w.z * pl[o.z] + w.w * pl[o.w];
        float f = v * cab[c];
        sum += f;
        mx = fmaxf(mx, f);
    }
    cat[(size_t)b * 2 * HW_ + pix]       = sum * (1.f / (float)C_);
    cat[(size_t)b * 2 * HW_ + HW_ + pix] = mx;
}

// ---------------------------------------------------------------------------
// sa = sigmoid(conv7x7(cat)), pad 3, 2 in-ch -> 1 out-ch.
// ---------------------------------------------------------------------------
__global__ __launch_bounds__(256) void conv_kernel(const float* __restrict__ cat,
                                                   const float* __restrict__ w_sp,
                                                   float* __restrict__ sa) {
    __shared__ float wk[2 * 49];
    if (threadIdx.x < 98) wk[threadIdx.x] = w_sp[threadIdx.x];
    __syncthreads();
    int t = blockIdx.x * blockDim.x + threadIdx.x;
    if (t >= B_ * HW_) return;
    int b = t / HW_, pix = t % HW_;
    int y = pix / W_, x = pix % W_;
    float acc = 0.f;
#pragma unroll
    for (int i = 0; i < 2; ++i) {
        const float* ch = cat + (size_t)b * 2 * HW_ + (size_t)i * HW_;
        for (int ky = 0; ky < 7; ++ky) {
            int yy = y + ky - 3;
            if (yy < 0 || yy >= H_) continue;
            for (int kx = 0; kx < 7; ++kx) {
                int xx = x + kx - 3;
                if (xx < 0 || xx >= W_) continue;
                acc += ch[yy * W_ + xx] * wk[i * 49 + ky * 7 + kx];
            }
        }
    }
    sa[t] = 1.f / (1.f + expf(-acc));
}

// ---------------------------------------------------------------------------
// fused += 0.25 * rotate_inv(fatt), fatt = rotate(x)*ca*sa — double gather via
// tables; x is L2-resident (134 MB < 192 MB L2), so no 134 MB intermediate.
// Table fetches are b128 (int4/float4).
// ---------------------------------------------------------------------------
__global__ __launch_bounds__(256) void final_kernel(const float* __restrict__ x,
                                                    const float* __restrict__ ca,
                                                    const float* __restrict__ sa,
                                                    const int4* __restrict__ F_off,
                                                    const float4* __restrict__ F_w,
                                                    const int4* __restrict__ I_idx,
                                                    const float4* __restrict__ I_w,
                                                    float* __restrict__ out,
                                                    int accumulate) {
    size_t t = (size_t)blockIdx.x * blockDim.x + threadIdx.x;
    if (t >= (size_t)B_ * C_ * HW_) return;
    size_t pix = t % HW_;
    size_t bc  = t / HW_;
    int c = (int)(bc % C_);
    int b = (int)(bc / C_);
    const float* plane = x + bc * HW_;
    const float* sab   = sa + (size_t)b * HW_;
    float cav = ca[b * C_ + c];

    int4   ii = I_idx[pix];
    float4 iw = I_w[pix];
    int   pj4[4] = {ii.x, ii.y, ii.z, ii.w};
    float wj4[4] = {iw.x, iw.y, iw.z, iw.w};

    float contrib = 0.f;
#pragma unroll
    for (int j = 0; j < 4; ++j) {
        int pj = pj4[j];
        int4   fo = F_off[pj];
        float4 fw = F_w[pj];
        float v = fw.x * plane[fo.x] + fw.y * plane[fo.y] +
                  fw.z * plane[fo.z] + fw.w * plane[fo.w];
        contrib += wj4[j] * v * sab[pj];
    }
    float r = contrib * cav * 0.25f;
    out[t] = accumulate ? (out[t] + r) : r;
}

// ---------------------------------------------------------------------------
extern "C" void kernel_launch(void* const* d_in, const int* in_sizes, int n_in,
                              void* d_out, int out_size, void* d_ws, size_t ws_size,
                              hipStream_t stream) {
    const float* x        = (const float*)d_in[0];
    const float* angles   = (const float*)d_in[1];
    const float* w1       = (const float*)d_in[2];
    const float* bn_gamma = (const float*)d_in[3];
    const float* bn_beta  = (const float*)d_in[4];
    const float* bn_mean  = (const float*)d_in[5];
    const float* bn_var   = (const float*)d_in[6];
    const float* w2       = (const float*)d_in[7];
    const float* w_sp     = (const float*)d_in[8];
    float* out = (float*)d_out;

    // workspace layout (all 256B-aligned by construction)
    char* ws = (char*)d_ws;
    size_t off = 0;
    auto take = [&](size_t bytes) { char* q = ws + off; off += (bytes + 255) & ~(size_t)255; return q; };
    int4*   F_off = (int4*)  take((size_t)HW_ * sizeof(int4));
    float4* F_w   = (float4*)take((size_t)HW_ * sizeof(float4));
    int4*   I_idx = (int4*)  take((size_t)HW_ * sizeof(int4));
    float4* I_w   = (float4*)take((size_t)HW_ * sizeof(float4));
    float*  Wmap  = (float*) take((size_t)HW_ * sizeof(float));
    float*  p     = (float*) take((size_t)B_ * C_ * sizeof(float));
    float*  ca    = (float*) take((size_t)B_ * C_ * sizeof(float));
    float*  cat   = (float*) take((size_t)B_ * 2 * HW_ * sizeof(float));
    float*  sa    = (float*) take((size_t)B_ * HW_ * sizeof(float));
    (void)ws_size; (void)in_sizes; (void)n_in; (void)out_size;

    const int TB = 256;
    dim3 gHW((HW_ + TB - 1) / TB);
    dim3 gBHW((B_ * HW_ + TB - 1) / TB);
    dim3 gAll((unsigned)(((size_t)B_ * C_ * HW_ + TB - 1) / TB));

    for (int ang = 0; ang < NANG_; ++ang) {
        zero_wmap_kernel<<<gHW, TB, 0, stream>>>(Wmap);
        build_tables_kernel<<<gHW, TB, 0, stream>>>(angles, ang, F_off, F_w, I_idx, I_w, Wmap);
        pool_kernel<<<dim3(B_ * C_), TB, 0, stream>>>(x, Wmap, p);
        fc_kernel<<<dim3(1), 512, 0, stream>>>(p, w1, bn_gamma, bn_beta, bn_mean, bn_var, w2, ca);
        cat_kernel<<<gBHW, TB, 0, stream>>>(x, ca, F_off, F_w, cat);
        conv_kernel<<<gBHW, TB, 0, stream>>>(cat, w_sp, sa);
        final_kernel<<<gAll, TB, 0, stream>>>(x, ca, sa, F_off, F_w, I_idx, I_w, out, ang == 0 ? 0 : 1);
    }
}